// LSHAttention_26139170963882
// MI455X (gfx1250) — compile-verified
//
#include <hip/hip_runtime.h>
#include <hip/hip_bf16.h>

// ---------------------------------------------------------------------------
// LSH attention for MI455X (gfx1250, wave32). f32 WMMA (V_WMMA_F32_16X16X4_F32)
// for projection, QK^T and PV GEMMs. Counting sort over 1024 buckets/batch.
// Two-pass attention (LSE pass + accumulate pass), ~6.5 MB workspace.
// Async global->LDS copies (GLOBAL_LOAD_ASYNC_TO_LDS_B128, ASYNCcnt path) +
// transposed R tile so every WMMA operand is a contiguous 64-bit LDS load.
// ---------------------------------------------------------------------------

#define BB 8          // batch
#define SS 8192       // sequence
#define DD 64         // head dim
#define HH 8          // num hash rounds
#define BKT 64        // bucket_size
#define BN 128        // bucket_nums = SS/BKT
#define NBUCKET (HH * BN)      // 1024 buckets per batch
#define CHUNKS (HH * SS / BKT) // 1024 chunks per batch

typedef float v2f __attribute__((ext_vector_type(2)));
typedef float v8f __attribute__((ext_vector_type(8)));

// ---------------------------------------------------------------------------
// Async global->LDS copy helpers (gfx1250 GLOBAL_LOAD_ASYNC_TO_LDS_B128).
// Builtin signature (from hipcc diagnostic): param0 = v4i addrspace(1)*,
// param1 = LDS dest, param2 = imm offset, param3 = cpol.
// Falls back to a synchronous 128-bit copy if the builtin is unavailable.
// ---------------------------------------------------------------------------
#if defined(__gfx1250__) && __has_builtin(__builtin_amdgcn_global_load_async_to_lds_b128)
#define LSH_ASYNC_CP 1
typedef int lsh_i32x4 __attribute__((vector_size(16)));
typedef lsh_i32x4 __attribute__((address_space(1))) lsh_g_v4i;
typedef lsh_i32x4 __attribute__((address_space(3))) lsh_l_v4i;
#endif

__device__ __forceinline__ void lsh_cp_b128(const float* g, float* l) {
#if defined(LSH_ASYNC_CP)
    __builtin_amdgcn_global_load_async_to_lds_b128(
        (lsh_g_v4i*)g, (lsh_l_v4i*)l, 0 /*offset*/, 0 /*cpol*/);
#else
    *(float4*)l = *(const float4*)g;
#endif
}

__device__ __forceinline__ void lsh_cp_wait() {
#if defined(LSH_ASYNC_CP)
#if __has_builtin(__builtin_amdgcn_s_wait_asynccnt)
    __builtin_amdgcn_s_wait_asynccnt(0);
#else
    asm volatile("s_wait_asynccnt 0" ::: "memory");
#endif
#endif
}

// ---------------------------------------------------------------------------
// generic zeroing kernels
// ---------------------------------------------------------------------------
__global__ void zero_f32_kernel(float* __restrict__ p) {
    p[(size_t)blockIdx.x * blockDim.x + threadIdx.x] = 0.0f;
}
__global__ void zero_i32_kernel(int* __restrict__ p) {
    p[(size_t)blockIdx.x * blockDim.x + threadIdx.x] = 0;
}

// ---------------------------------------------------------------------------
// Kernel 1: LSH projection + argmax bucket assignment.
// One block = 64 query rows of one (b,h). xR = Q(64x64) * R_bh(64x64) via WMMA,
// then per-row argmax over [xR, -xR] (128 candidate buckets).
// R is staged TRANSPOSED in LDS so B operands are contiguous k-pairs.
// ---------------------------------------------------------------------------
__global__ __launch_bounds__(256) void lsh_proj_hash_kernel(
    const float* __restrict__ qk, const float* __restrict__ Rm,
    int* __restrict__ bws, int* __restrict__ bout)
{
    __shared__ float sRt[64 * 64];  // R^T: sRt[n][k]  16KB
    __shared__ float sQ[64 * 64];   // Q rows          16KB
    __shared__ float sX[64 * 64];   // xR result       16KB

    const int blk  = blockIdx.x;
    const int nMB  = SS / 64;              // 128 row-blocks per (b,h)
    const int mb   = blk % nMB;
    const int h    = (blk / nMB) % HH;
    const int b    = blk / (nMB * HH);
    const int t0   = mb * 64;
    const int tid  = threadIdx.x;

    // Q tile: contiguous rows -> async B128 segments (64 rows x 16 segs)
    for (int i = tid; i < 64 * 16; i += 256) {
        const int m = i >> 4, s4 = (i & 15) * 4;
        lsh_cp_b128(&qk[((size_t)b * SS + t0 + m) * DD + s4], &sQ[m * 64 + s4]);
    }
    // R^T tile: transpose on store (LDS writes conflict-free along f)
    for (int i = tid; i < 64 * 64; i += 256) {
        const int r = i >> 6, f = i & 63;
        sRt[r * 64 + f] = Rm[(((size_t)b * DD + f) * HH + h) * 64 + r]; // R[b,f,h,r]
    }
    lsh_cp_wait();
    __syncthreads();

    const int wave = tid >> 5, lane = tid & 31;
    const int lm = lane & 15, kh = lane >> 4;
    const int mt  = wave >> 1;          // 4 m-tiles, 2 waves each
    const int nt0 = (wave & 1) * 2;     // each wave: 2 n-tiles

    v8f acc0 = {}; v8f acc1 = {};
    for (int k0 = 0; k0 < 64; k0 += 4) {
        v2f a;  a.x  = sQ[(mt * 16 + lm) * 64 + k0 + 2 * kh];
                a.y  = sQ[(mt * 16 + lm) * 64 + k0 + 2 * kh + 1];
        v2f b0; b0.x = sRt[(nt0 * 16 + lm) * 64 + k0 + 2 * kh];
                b0.y = sRt[(nt0 * 16 + lm) * 64 + k0 + 2 * kh + 1];
        v2f b1; b1.x = sRt[((nt0 + 1) * 16 + lm) * 64 + k0 + 2 * kh];
                b1.y = sRt[((nt0 + 1) * 16 + lm) * 64 + k0 + 2 * kh + 1];
        acc0 = __builtin_amdgcn_wmma_f32_16x16x4_f32(false, a, false, b0, (short)0, acc0, false, false);
        acc1 = __builtin_amdgcn_wmma_f32_16x16x4_f32(false, a, false, b1, (short)0, acc1, false, false);
    }
    for (int j = 0; j < 8; ++j) {
        const int m = mt * 16 + j + 8 * kh;
        sX[m * 64 + nt0 * 16 + lm]       = acc0[j];
        sX[m * 64 + (nt0 + 1) * 16 + lm] = acc1[j];
    }
    __syncthreads();

    if (tid < 64) {
        float best = sX[tid * 64 + 0];
        int   bi   = 0;
        for (int j = 1; j < 128; ++j) {
            const float vv = (j < 64) ? sX[tid * 64 + j] : -sX[tid * 64 + (j - 64)];
            if (vv > best) { best = vv; bi = j; }
        }
        const int    bkt = bi + h * BN;
        const size_t e   = (size_t)b * (HH * SS) + (size_t)h * SS + (t0 + tid);
        bws[e]  = bkt;
        bout[e] = bkt;   // "buckets" tuple output
    }
}

// ---------------------------------------------------------------------------
// Kernel 2: per-batch bucket histogram (atomic).
// ---------------------------------------------------------------------------
__global__ __launch_bounds__(256) void lsh_hist_kernel(
    const int* __restrict__ buckets, int* __restrict__ hist)
{
    const size_t i = (size_t)blockIdx.x * blockDim.x + threadIdx.x;  // B*H*S
    const int b = (int)(i / (HH * SS));
    atomicAdd(&hist[b * NBUCKET + buckets[i]], 1);
}

// ---------------------------------------------------------------------------
// Kernel 3: exclusive scan of 1024 bucket counts per batch.
// ---------------------------------------------------------------------------
__global__ void lsh_scan_kernel(const int* __restrict__ hist, int* __restrict__ offs)
{
    if (threadIdx.x == 0) {
        const int b = blockIdx.x;
        int acc = 0;
        for (int i = 0; i < NBUCKET; ++i) {
            offs[b * NBUCKET + i] = acc;
            acc += hist[b * NBUCKET + i];
        }
    }
}

// ---------------------------------------------------------------------------
// Kernel 4: stable scatter. One wave32 per (b,h) row walks t in order;
// intra-wave stable ranks via lane matching. Buckets of distinct h are
// disjoint, so per-(b,h) local counters suffice.
// ---------------------------------------------------------------------------
__global__ __launch_bounds__(32) void lsh_scatter_kernel(
    const int* __restrict__ buckets, const int* __restrict__ offs,
    int* __restrict__ pos_sorted)
{
    __shared__ int cnt[BN];
    const int b = blockIdx.x / HH, h = blockIdx.x % HH;
    const int lane = threadIdx.x;
    for (int i = lane; i < BN; i += 32) cnt[i] = 0;
    __syncthreads();

    const int* bk  = buckets + (size_t)b * (HH * SS) + (size_t)h * SS;
    const int* off = offs + b * NBUCKET;
    int* out = pos_sorted + (size_t)b * (HH * SS);

    for (int t0 = 0; t0 < SS; t0 += 32) {
        const int t     = t0 + lane;
        const int bkt   = bk[t];
        const int local = bkt - h * BN;      // [0,128)
        unsigned mask = 0u;
        for (int j = 0; j < 32; ++j) {
            const int oj = __shfl(local, j, 32);
            mask |= (oj == local) ? (1u << j) : 0u;
        }
        const unsigned lower = mask & ((1u << lane) - 1u);
        const int rank = __popc(lower);
        const int prev = cnt[local];
        __syncthreads();
        if (lower == 0u) cnt[local] = prev + __popc(mask);   // group leader
        __syncthreads();
        out[off[bkt] + prev + rank] = t;
    }
}

// ---------------------------------------------------------------------------
// Kernel 5/7: chunked attention. One block per (b, chunk). Async-gathers Q
// (64 rows) and K (128 rows: chunk + prev chunk, wrapped) into LDS, normalizes
// K, computes scores via WMMA f32 16x16x4, masks self-attention, row-softmax
// stats. WITH_PV=false: write per-row LSE. WITH_PV=true: P = exp(S - lse),
// PV via WMMA with V streamed from L2, scatter-add weighted rows into out.
// ---------------------------------------------------------------------------
template <bool WITH_PV>
__global__ __launch_bounds__(256) void lsh_attn_kernel(
    const float* __restrict__ qk, const float* __restrict__ vg,
    const int* __restrict__ pos, float* __restrict__ lse_g,
    const float* __restrict__ lse_tot, float* __restrict__ outp)
{
    __shared__ __align__(16) float uni[12288]; // 48KB union
    float* sQ = uni;             // [64][64]   phase 1
    float* sK = uni + 64 * 64;   // [128][64]  phase 1
    float* sP = uni;             // [64][128]  phase 2 (overlaps sQ/sK)
    __shared__ int   q_pos[64];
    __shared__ int   kv_pos[128];
    __shared__ float rnorm[128];
    __shared__ float sLse[64];
    __shared__ float sW[64];

    const int blk = blockIdx.x;
    const int c   = blk % CHUNKS;
    const int b   = blk / CHUNKS;
    const int hh  = c >> 7;                 // hash round of this chunk
    const int tid = threadIdx.x;
    const size_t pbase = (size_t)b * (HH * SS);

    if (tid < 64) {
        const int p = pos[pbase + (size_t)c * 64 + tid];
        q_pos[tid] = p;  kv_pos[tid] = p;
    } else if (tid < 128) {
        const int pc = (c + CHUNKS - 1) % CHUNKS;
        kv_pos[tid] = pos[pbase + (size_t)pc * 64 + (tid - 64)];
    }
    __syncthreads();

    // async gather: Q rows (64x16 B128 segs) and K rows (128x16 B128 segs)
    const float* qb = qk + (size_t)b * SS * DD;
    for (int i = tid; i < 64 * 16; i += 256) {
        const int m = i >> 4, s4 = (i & 15) * 4;
        lsh_cp_b128(&qb[(size_t)q_pos[m] * DD + s4], &sQ[m * 64 + s4]);
    }
    for (int i = tid; i < 128 * 16; i += 256) {
        const int r = i >> 4, s4 = (i & 15) * 4;
        lsh_cp_b128(&qb[(size_t)kv_pos[r] * DD + s4], &sK[r * 64 + s4]);
    }
    lsh_cp_wait();
    __syncthreads();

    if (tid < 128) {
        float ss = 0.0f;
        for (int f = 0; f < 64; ++f) { const float x = sK[tid * 64 + f]; ss += x * x; }
        rnorm[tid] = rsqrtf(ss);
    }
    __syncthreads();
    for (int i = tid; i < 128 * 64; i += 256) sK[i] *= rnorm[i >> 6];
    __syncthreads();

    const int wave = tid >> 5, lane = tid & 31;
    const int lm = lane & 15, kh = lane >> 4;

    // scores: wave owns n-strip [16*wave, 16*wave+16) across all 4 m-tiles
    v8f sacc[4] = {};
    for (int k0 = 0; k0 < 64; k0 += 4) {
        v2f bb; bb.x = sK[(16 * wave + lm) * 64 + k0 + 2 * kh];
                bb.y = sK[(16 * wave + lm) * 64 + k0 + 2 * kh + 1];
        for (int mt2 = 0; mt2 < 4; ++mt2) {
            v2f a; a.x = sQ[(mt2 * 16 + lm) * 64 + k0 + 2 * kh];
                   a.y = sQ[(mt2 * 16 + lm) * 64 + k0 + 2 * kh + 1];
            sacc[mt2] = __builtin_amdgcn_wmma_f32_16x16x4_f32(false, a, false, bb, (short)0, sacc[mt2], false, false);
        }
    }
    __syncthreads();  // all waves done reading sQ/sK before overwrite

    for (int mt2 = 0; mt2 < 4; ++mt2)
        for (int j = 0; j < 8; ++j) {
            const int m = mt2 * 16 + j + 8 * kh;
            const int n = 16 * wave + lm;
            float sv = 0.125f * sacc[mt2][j];          // D^-0.5
            if (q_pos[m] == kv_pos[n]) sv = -1.0e5f;   // self mask
            sP[m * 128 + n] = sv;
        }
    __syncthreads();

    // row softmax stats: wave handles 8 rows
    for (int rr = 0; rr < 8; ++rr) {
        const int m = wave * 8 + rr;
        const float v0 = sP[m * 128 + lane];
        const float v1 = sP[m * 128 + lane + 32];
        const float v2 = sP[m * 128 + lane + 64];
        const float v3 = sP[m * 128 + lane + 96];
        float mx = fmaxf(fmaxf(v0, v1), fmaxf(v2, v3));
        for (int o = 16; o > 0; o >>= 1) mx = fmaxf(mx, __shfl_xor(mx, o, 32));
        float se = __expf(v0 - mx) + __expf(v1 - mx) + __expf(v2 - mx) + __expf(v3 - mx);
        for (int o = 16; o > 0; o >>= 1) se += __shfl_xor(se, o, 32);
        if (lane == 0) sLse[m] = mx + __logf(se);
    }
    __syncthreads();

    if (!WITH_PV) {
        if (tid < 64)
            lse_g[((size_t)b * HH + hh) * SS + q_pos[tid]] = sLse[tid];
        return;
    }

    if (tid < 64)
        sW[tid] = __expf(sLse[tid] - lse_tot[(size_t)b * SS + q_pos[tid]]);
    for (int i = tid; i < 64 * 128; i += 256) {
        const int m = i >> 7;
        sP[i] = __expf(sP[i] - sLse[m]);
    }
    __syncthreads();

    // PV: out(64x64) = P(64x128) x V(128x64); wave handles 2 n-tiles of one m-tile
    const int mt  = wave >> 1;
    const int nt0 = (wave & 1) * 2;
    v8f o0 = {}; v8f o1 = {};
    const float* vb = vg + (size_t)b * SS * DD;
    for (int k0 = 0; k0 < 128; k0 += 4) {
        v2f a; a.x = sP[(mt * 16 + lm) * 128 + k0 + 2 * kh];
               a.y = sP[(mt * 16 + lm) * 128 + k0 + 2 * kh + 1];
        const int r0 = kv_pos[k0 + 2 * kh];
        const int r1 = kv_pos[k0 + 2 * kh + 1];
        v2f b0; b0.x = vb[(size_t)r0 * DD + nt0 * 16 + lm];
                b0.y = vb[(size_t)r1 * DD + nt0 * 16 + lm];
        v2f b1; b1.x = vb[(size_t)r0 * DD + (nt0 + 1) * 16 + lm];
                b1.y = vb[(size_t)r1 * DD + (nt0 + 1) * 16 + lm];
        o0 = __builtin_amdgcn_wmma_f32_16x16x4_f32(false, a, false, b0, (short)0, o0, false, false);
        o1 = __builtin_amdgcn_wmma_f32_16x16x4_f32(false, a, false, b1, (short)0, o1, false, false);
    }
    for (int j = 0; j < 8; ++j) {
        const int   m   = mt * 16 + j + 8 * kh;
        const float wgt = sW[m];
        const int   t   = q_pos[m];
        atomicAdd(&outp[((size_t)b * SS + t) * DD + nt0 * 16 + lm],       wgt * o0[j]);
        atomicAdd(&outp[((size_t)b * SS + t) * DD + (nt0 + 1) * 16 + lm], wgt * o1[j]);
    }
}

// ---------------------------------------------------------------------------
// Kernel 6: cross-hash logsumexp of per-row LSEs.
// ---------------------------------------------------------------------------
__global__ __launch_bounds__(256) void lsh_lse_merge_kernel(
    const float* __restrict__ lse, float* __restrict__ lse_tot)
{
    const size_t i = (size_t)blockIdx.x * blockDim.x + threadIdx.x;  // B*S
    const int b = (int)(i / SS), t = (int)(i % SS);
    float vals[HH];
    float mx = -3.0e38f;
    for (int h = 0; h < HH; ++h) {
        vals[h] = lse[((size_t)b * HH + h) * SS + t];
        mx = fmaxf(mx, vals[h]);
    }
    float se = 0.0f;
    for (int h = 0; h < HH; ++h) se += __expf(vals[h] - mx);
    lse_tot[i] = mx + __logf(se);
}

// ---------------------------------------------------------------------------
// launch
// ---------------------------------------------------------------------------
extern "C" void kernel_launch(void* const* d_in, const int* in_sizes, int n_in,
                              void* d_out, int out_size, void* d_ws, size_t ws_size,
                              hipStream_t stream) {
    (void)in_sizes; (void)n_in; (void)out_size; (void)ws_size;
    const float* qk = (const float*)d_in[0];
    const float* vv = (const float*)d_in[1];
    const float* Rm = (const float*)d_in[2];

    float* out  = (float*)d_out;                      // [B,S,D] f32
    int*   bout = (int*)(out + (size_t)BB * SS * DD); // buckets [B, H*S]

    char* w = (char*)d_ws;
    int*   ws_buckets = (int*)(w);                       // 2 MB
    int*   ws_hist    = (int*)(w + 2097152);             // 32 KB
    int*   ws_offs    = (int*)(w + 2129920);             // 32 KB
    int*   ws_pos     = (int*)(w + 2162688);             // 2 MB
    float* ws_lse     = (float*)(w + 4259840);           // 2 MB
    float* ws_lsetot  = (float*)(w + 6356992);           // 256 KB

    zero_f32_kernel<<<(BB * SS * DD) / 256, 256, 0, stream>>>(out);
    zero_i32_kernel<<<(BB * NBUCKET) / 256, 256, 0, stream>>>(ws_hist);

    lsh_proj_hash_kernel<<<BB * HH * (SS / 64), 256, 0, stream>>>(qk, Rm, ws_buckets, bout);
    lsh_hist_kernel<<<(BB * HH * SS) / 256, 256, 0, stream>>>(ws_buckets, ws_hist);
    lsh_scan_kernel<<<BB, 32, 0, stream>>>(ws_hist, ws_offs);
    lsh_scatter_kernel<<<BB * HH, 32, 0, stream>>>(ws_buckets, ws_offs, ws_pos);

    lsh_attn_kernel<false><<<BB * CHUNKS, 256, 0, stream>>>(qk, vv, ws_pos, ws_lse, ws_lsetot, out);
    lsh_lse_merge_kernel<<<(BB * SS) / 256, 256, 0, stream>>>(ws_lse, ws_lsetot);
    lsh_attn_kernel<true><<<BB * CHUNKS, 256, 0, stream>>>(qk, vv, ws_pos, ws_lse, ws_lsetot, out);
}